// GNN_60885456389012
// MI455X (gfx1250) — compile-verified
//
#include <hip/hip_runtime.h>
#include <hip/hip_bf16.h>

typedef __attribute__((ext_vector_type(2))) float v2f;
typedef __attribute__((ext_vector_type(8))) float v8f;

#define N_NODES  100000
#define N_EDGES  1600000
#define HIDDEN   64
#define N_GRAPHS 64
#define NODE_F   (N_NODES * HIDDEN)          // 6,400,000 floats
#define NEG_FLT_MAX (-3.402823466e38f)

// ---------------------------------------------------------------- fill kernels
__global__ void gnn_fill_zero4(float4* __restrict__ p, int n4) {
    int t = blockIdx.x * blockDim.x + threadIdx.x;
    if (t < n4) p[t] = make_float4(0.f, 0.f, 0.f, 0.f);
}

__global__ void gnn_fill_neginf(float* __restrict__ p, int n) {
    int t = blockIdx.x * blockDim.x + threadIdx.x;
    if (t < n) p[t] = NEG_FLT_MAX;
}

// ---------------------------------------------------------------- edge scatter
// 16 lanes per edge, each lane handles 4 features (float4 gather, 4 f32 atomics)
__global__ void gnn_scatter_add(const float* __restrict__ h,
                                const int* __restrict__ src,
                                const int* __restrict__ dst,
                                float* __restrict__ agg) {
    long long t = (long long)blockIdx.x * blockDim.x + threadIdx.x;
    int e  = (int)(t >> 4);
    int f4 = (int)(t & 15);
    if (e >= N_EDGES) return;
    int s = src[e];
    int d = dst[e];
    const float4 v = *(const float4*)(h + (long long)s * HIDDEN + f4 * 4);
    float* o = agg + (long long)d * HIDDEN + f4 * 4;
    atomicAdd(o + 0, v.x);
    atomicAdd(o + 1, v.y);
    atomicAdd(o + 2, v.z);
    atomicAdd(o + 3, v.w);
}

// ---------------------------------------------------------------- WMMA GEMM
// out[N,64] = maybe_relu( agg @ w_rel + xin @ w_root + b )
// grid.x = N/16 (6250, exact), block = 128 (4 waves); wave w owns cols [16w,16w+16)
__global__ __launch_bounds__(128)
void gnn_layer_gemm(const float* __restrict__ agg,
                    const float* __restrict__ xin,
                    const float* __restrict__ w_rel,
                    const float* __restrict__ w_root,
                    const float* __restrict__ bias,
                    float* __restrict__ out,
                    int do_relu) {
    const int lane    = threadIdx.x & 31;
    const int wave    = threadIdx.x >> 5;        // output col tile 0..3
    const int row0    = blockIdx.x * 16;
    const int colBase = wave * 16;

    const int m  = lane & 15;     // A row within tile / D column selector
    const int kh = lane >> 4;     // 0/1: K-half for A/B, row-half for D
    const int n  = lane & 15;     // B/D column within tile

    v8f c = {};                   // 16x16 f32 accumulator (8 VGPRs)

    #pragma unroll
    for (int k0 = 0; k0 < HIDDEN; k0 += 4) {
        const int ka = k0 + 2 * kh;
        // ---- agg @ w_rel
        v2f a1 = *(const v2f*)(agg + (long long)(row0 + m) * HIDDEN + ka);
        v2f b1;
        b1.x = w_rel[(ka + 0) * HIDDEN + colBase + n];
        b1.y = w_rel[(ka + 1) * HIDDEN + colBase + n];
        c = __builtin_amdgcn_wmma_f32_16x16x4_f32(false, a1, false, b1,
                                                  (short)0, c, false, false);
        // ---- xin @ w_root
        v2f a2 = *(const v2f*)(xin + (long long)(row0 + m) * HIDDEN + ka);
        v2f b2;
        b2.x = w_root[(ka + 0) * HIDDEN + colBase + n];
        b2.y = w_root[(ka + 1) * HIDDEN + colBase + n];
        c = __builtin_amdgcn_wmma_f32_16x16x4_f32(false, a2, false, b2,
                                                  (short)0, c, false, false);
    }

    const float bv = bias[colBase + n];
    #pragma unroll
    for (int r = 0; r < 8; ++r) {
        int row = row0 + r + 8 * kh;          // D layout: lanes 16-31 hold M=8..15
        float v = c[r] + bv;
        if (do_relu) v = fmaxf(v, 0.f);
        out[(long long)row * HIDDEN + colBase + n] = v;
    }
}

// ---------------------------------------------------------------- segment max
__device__ inline void atomicMaxFloat(float* addr, float val) {
    if (val >= 0.f)
        atomicMax((int*)addr, __float_as_int(val));
    else
        atomicMin((unsigned int*)addr, __float_as_uint(val));
}

__global__ void gnn_seg_max(const float* __restrict__ h,
                            const int* __restrict__ batch,
                            float* __restrict__ g) {
    long long t = (long long)blockIdx.x * blockDim.x + threadIdx.x;
    int node = (int)(t >> 4);
    int f4   = (int)(t & 15);
    if (node >= N_NODES) return;
    int b = batch[node];
    const float4 v = *(const float4*)(h + (long long)node * HIDDEN + f4 * 4);
    float* o = g + (long long)b * HIDDEN + f4 * 4;
    atomicMaxFloat(o + 0, v.x);
    atomicMaxFloat(o + 1, v.y);
    atomicMaxFloat(o + 2, v.z);
    atomicMaxFloat(o + 3, v.w);
}

// ---------------------------------------------------------------- tiny MLP head
__global__ void gnn_mlp(const float* __restrict__ g,
                        const float* __restrict__ w1, const float* __restrict__ b1,
                        const float* __restrict__ w2, const float* __restrict__ b2,
                        float* __restrict__ out) {
    int i = threadIdx.x;              // graph id, 64 threads
    if (i >= N_GRAPHS) return;
    float hid[5];
    #pragma unroll
    for (int j = 0; j < 5; ++j) {
        float acc = b1[j];
        for (int k = 0; k < HIDDEN; ++k)
            acc += g[i * HIDDEN + k] * w1[k * 5 + j];
        hid[j] = fmaxf(acc, 0.f);
    }
    float o = b2[0];
    #pragma unroll
    for (int j = 0; j < 5; ++j) o += hid[j] * w2[j];
    out[i] = o;
}

// ---------------------------------------------------------------- orchestration
extern "C" void kernel_launch(void* const* d_in, const int* in_sizes, int n_in,
                              void* d_out, int out_size, void* d_ws, size_t ws_size,
                              hipStream_t stream) {
    const float* x       = (const float*)d_in[0];
    const int*   eidx    = (const int*)d_in[1];
    const int*   batch   = (const int*)d_in[2];
    const float* w_rel1  = (const float*)d_in[3];
    const float* w_root1 = (const float*)d_in[4];
    const float* b1      = (const float*)d_in[5];
    const float* w_rel2  = (const float*)d_in[6];
    const float* w_root2 = (const float*)d_in[7];
    const float* b2      = (const float*)d_in[8];
    const float* w_rel3  = (const float*)d_in[9];
    const float* w_root3 = (const float*)d_in[10];
    const float* b3      = (const float*)d_in[11];
    const float* w_mlp1  = (const float*)d_in[12];
    const float* b_mlp1  = (const float*)d_in[13];
    const float* w_mlp2  = (const float*)d_in[14];
    const float* b_mlp2  = (const float*)d_in[15];
    float* out = (float*)d_out;

    const int* src = eidx;             // edge_index[0]
    const int* dst = eidx + N_EDGES;   // edge_index[1]

    float* ws  = (float*)d_ws;
    float* agg = ws;                   // [N_NODES, 64]
    float* hA  = ws + (size_t)NODE_F;  // [N_NODES, 64]
    float* hB  = hA + (size_t)NODE_F;  // [N_NODES, 64]
    float* g   = hB + (size_t)NODE_F;  // [N_GRAPHS, 64]

    const int zeroBlocks    = (NODE_F / 4 + 255) / 256;          // float4 fill
    const int scatterBlocks = (int)(((long long)N_EDGES * 16 + 255) / 256);
    const int gemmBlocks    = N_NODES / 16;                      // 6250, exact
    const int segBlocks     = (int)(((long long)N_NODES * 16 + 255) / 256);

    const float* in   = x;
    float*       outs[3] = { hA, hB, hA };
    const float* wr[3]   = { w_rel1,  w_rel2,  w_rel3  };
    const float* wo[3]   = { w_root1, w_root2, w_root3 };
    const float* bb[3]   = { b1, b2, b3 };

    for (int layer = 0; layer < 3; ++layer) {
        gnn_fill_zero4<<<zeroBlocks, 256, 0, stream>>>((float4*)agg, NODE_F / 4);
        gnn_scatter_add<<<scatterBlocks, 256, 0, stream>>>(in, src, dst, agg);
        gnn_layer_gemm<<<gemmBlocks, 128, 0, stream>>>(
            agg, in, wr[layer], wo[layer], bb[layer], outs[layer],
            layer < 2 ? 1 : 0);
        in = outs[layer];
    }

    // global max pool:  g = segment_max(h3, batch)
    gnn_fill_neginf<<<(N_GRAPHS * HIDDEN + 255) / 256, 256, 0, stream>>>(
        g, N_GRAPHS * HIDDEN);
    gnn_seg_max<<<segBlocks, 256, 0, stream>>>(in, batch, g);

    // MLP head -> [64,1]
    gnn_mlp<<<1, 64, 0, stream>>>(g, w_mlp1, b_mlp1, w_mlp2, b_mlp2, out);
}